// SEBlock_35519379537932
// MI455X (gfx1250) — compile-verified
//
#include <hip/hip_runtime.h>

typedef __attribute__((ext_vector_type(2))) float v2f;
typedef __attribute__((ext_vector_type(4))) float v4f;
typedef __attribute__((ext_vector_type(8))) float v8f;

#define C_CH 256
#define R_RED 16
#define HW   3136   // 56*56
#define HW4  784    // float4 per channel plane
#define NB   32

// ---------------------------------------------------------------------------
// Kernel 1: squeeze (global average pool). One 256-thread block per (b,c)
// plane; b128 loads keep HBM busy and populate L2 with x (RT hint).
// ---------------------------------------------------------------------------
__global__ void se_pool(const float* __restrict__ x, float* __restrict__ s) {
  const unsigned plane = blockIdx.x;                 // b*C + c, 0..8191
  const v4f* p4 = (const v4f*)(x + (size_t)plane * HW);
  const int t = threadIdx.x;                          // 0..255
  float acc = 0.0f;
#pragma unroll
  for (int i = 0; i < 3; ++i) {                       // 3*256 = 768 vec4
    v4f v = p4[t + i * 256];
    acc += v.x + v.y + v.z + v.w;
  }
  if (t < (HW4 - 768)) {                              // 16-vec4 tail
    v4f v = p4[768 + t];
    acc += v.x + v.y + v.z + v.w;
  }
  __shared__ float red[256];
  red[t] = acc;
  __syncthreads();
#pragma unroll
  for (int off = 128; off > 0; off >>= 1) {
    if (t < off) red[t] += red[t + off];
    __syncthreads();
  }
  if (t == 0) s[plane] = red[0] * (1.0f / (float)HW);
}

// ---------------------------------------------------------------------------
// Kernel 2: excitation MLP on the WMMA path.
//   h    = relu(s @ w1^T)        (32x256)x(256x16) -> 32x16
//   gate = sigmoid(h @ w2^T)     (32x16)x(16x256)  -> 32x256
// 1 block, 2 waves; wave w owns batch rows 16w..16w+15.
// V_WMMA_F32_16X16X4_F32:
//   A (16x4): lanes 0-15 -> {K0,K1}, lanes 16-31 -> {K2,K3}, M = lane&15
//   B (4x16): same striping with N = lane&15
//   C/D (16x16): vgpr r, lane L -> M = r + 8*(L>>4), N = L&15
// ---------------------------------------------------------------------------
__global__ void se_mlp(const float* __restrict__ s,
                       const float* __restrict__ w1,   // (16,256) row-major
                       const float* __restrict__ w2,   // (256,16) row-major
                       float* __restrict__ gate) {     // (32,256)
  const int lane = threadIdx.x & 31;
  const int wv   = threadIdx.x >> 5;       // 0..1  (M tile)
  const int half = lane >> 4;              // 0/1   (K sub-group)
  const int ml   = lane & 15;              // M row (A) / N col (B)

  __shared__ float hbuf[2][16 * 16];

  // ---- GEMM1: accumulate K=256 in 64 WMMA steps ----
  v8f acc = {};
  const float* sA = s  + (size_t)(16 * wv + ml) * C_CH + 2 * half; // s row
  const float* sB = w1 + (size_t)ml * C_CH + 2 * half;             // w1 row (=B col)
#pragma unroll 8
  for (int k0 = 0; k0 < C_CH; k0 += 4) {
    v2f a = *(const v2f*)(sA + k0);
    v2f b = *(const v2f*)(sB + k0);
    acc = __builtin_amdgcn_wmma_f32_16x16x4_f32(false, a, false, b,
                                                (short)0, acc, false, false);
  }

  // ReLU + write h tile (D layout -> row major in LDS)
  {
    const int n     = lane & 15;
    const int mbase = 8 * (lane >> 4);
#pragma unroll
    for (int r = 0; r < 8; ++r) {
      float hv = acc[r];
      hbuf[wv][(mbase + r) * 16 + n] = hv > 0.0f ? hv : 0.0f;
    }
  }
  __syncthreads();

  // ---- GEMM2: 16 N-tiles, K=16 in 4 WMMA steps each ----
  const float* hA = &hbuf[wv][0];
  for (int nt = 0; nt < 16; ++nt) {
    v8f c2 = {};
#pragma unroll
    for (int k0 = 0; k0 < 16; k0 += 4) {
      v2f a = *(const v2f*)(hA + ml * 16 + k0 + 2 * half);
      v2f b = *(const v2f*)(w2 + (size_t)(16 * nt + ml) * R_RED + k0 + 2 * half);
      c2 = __builtin_amdgcn_wmma_f32_16x16x4_f32(false, a, false, b,
                                                 (short)0, c2, false, false);
    }
    const int n     = lane & 15;
    const int mbase = 8 * (lane >> 4);
#pragma unroll
    for (int r = 0; r < 8; ++r) {
      float g = 1.0f / (1.0f + __expf(-c2[r]));
      gate[(size_t)(16 * wv + mbase + r) * C_CH + 16 * nt + n] = g;
    }
  }
}

// ---------------------------------------------------------------------------
// Kernel 3: scale. x is resident in the 192MB L2 after se_pool; read it NT
// (last use) and store the output NT so the 103MB output stream does not
// evict x while this pass runs.
// ---------------------------------------------------------------------------
__global__ void se_scale(const float* __restrict__ x,
                         const float* __restrict__ gate,
                         float* __restrict__ out) {
  const unsigned i4    = blockIdx.x * blockDim.x + threadIdx.x; // 0..6422527
  const unsigned plane = i4 / HW4;                              // b*C + c
  const float g = gate[plane];
  v4f v = __builtin_nontemporal_load(((const v4f*)x) + i4);
  v4f r = v * g;
  __builtin_nontemporal_store(r, ((v4f*)out) + i4);
}

// ---------------------------------------------------------------------------
extern "C" void kernel_launch(void* const* d_in, const int* in_sizes, int n_in,
                              void* d_out, int out_size, void* d_ws, size_t ws_size,
                              hipStream_t stream) {
  const float* x  = (const float*)d_in[0];   // (32,256,56,56)
  const float* w1 = (const float*)d_in[1];   // (16,256)
  const float* w2 = (const float*)d_in[2];   // (256,16)
  float* out  = (float*)d_out;

  float* s    = (float*)d_ws;                // 8192 floats
  float* gate = s + NB * C_CH;               // 8192 floats

  se_pool <<<NB * C_CH, 256, 0, stream>>>(x, s);
  se_mlp  <<<1, 64, 0, stream>>>(s, w1, w2, gate);
  const int nvec4 = NB * C_CH * (HW / 4);    // 6,422,528
  se_scale<<<nvec4 / 256, 256, 0, stream>>>(x, gate, out);
}